// RoIHeads_80298708566706
// MI455X (gfx1250) — compile-verified
//
#include <hip/hip_runtime.h>
#include <math.h>

// ---------------- problem constants (match reference) ----------------
#define B_        4
#define N_        4000
#define C_        91
#define M_        (N_ * (C_ - 1))      // 360000 candidates / image
#define DETS_     100
#define WIMG      800.0f
#define SCORE_T_  0.05f
#define NMS_T_    0.5f
#define MIN_SZ_   0.01f
#define NEGV      (-1.0e30f)
#define NEGH      (-5.0e29f)           // NEG * 0.5 threshold from reference
#define BBOX_CLIP 4.135166556742356f   // log(1000/16)

// ---------------- NMS tiling ----------------
#define BPI   16                       // persistent blocks per image
#define TPB   256                      // 8 waves (wave32)
#define CHUNK (M_ / BPI)               // 22500 scores per block, LDS-resident (90 KB)
#define GRP   (CHUNK / 4)              // 5625 float4 groups per block

// ------------- CDNA5 async global->LDS (ASYNCcnt path) -------------
typedef int v4i __attribute__((vector_size(16)));
typedef __attribute__((address_space(1))) v4i* g_v4i_p;   // global (AS1)
typedef __attribute__((address_space(3))) v4i* l_v4i_p;   // LDS (AS3)

__device__ __forceinline__ void async_load_f4(const float* g, float* l) {
#if __has_builtin(__builtin_amdgcn_global_load_async_to_lds_b128)
  __builtin_amdgcn_global_load_async_to_lds_b128(
      (g_v4i_p)(unsigned long long)(const void*)g,
      (l_v4i_p)(unsigned)(unsigned long long)(void*)l,   // low 32 bits of flat = LDS offset
      0, 0);
#else
  unsigned lds_off = (unsigned)(unsigned long long)(void*)l;
  unsigned long long ga = (unsigned long long)(const void*)g;
  asm volatile("global_load_async_to_lds_b128 %0, %1, off"
               :: "v"(lds_off), "v"(ga) : "memory");
#endif
}

__device__ __forceinline__ void wait_async0() {
#if __has_builtin(__builtin_amdgcn_s_wait_asynccnt)
  __builtin_amdgcn_s_wait_asynccnt(0);
#else
  asm volatile("s_wait_asynccnt 0" ::: "memory");
#endif
}

// ------------- cross-block (per-image) sense-reversing barrier -------------
__device__ __forceinline__ void image_barrier(unsigned* cnt, unsigned* gen, int img) {
  __syncthreads();
  if (threadIdx.x == 0) {
    __threadfence();
    unsigned g = __hip_atomic_load(&gen[img], __ATOMIC_RELAXED, __HIP_MEMORY_SCOPE_AGENT);
    unsigned prev = __hip_atomic_fetch_add(&cnt[img], 1u, __ATOMIC_ACQ_REL,
                                           __HIP_MEMORY_SCOPE_AGENT);
    if (prev == BPI - 1) {
      __hip_atomic_store(&cnt[img], 0u, __ATOMIC_RELAXED, __HIP_MEMORY_SCOPE_AGENT);
      __hip_atomic_store(&gen[img], g + 1u, __ATOMIC_RELEASE, __HIP_MEMORY_SCOPE_AGENT);
    } else {
      while (__hip_atomic_load(&gen[img], __ATOMIC_ACQUIRE, __HIP_MEMORY_SCOPE_AGENT) == g)
        __builtin_amdgcn_s_sleep(2);
    }
  }
  __syncthreads();
}

// ------------- phase 2: persistent cooperative greedy NMS -------------
// Scores LDS-resident (filled via async global->LDS, ASYNCcnt).
// Per round: fused suppression + argmax sweep over float4 LDS groups,
// packed-u64 wave32 shuffle reduction, image-wide atomicMax, 2 spin barriers.
// Class-offset shortcut: different labels can never overlap => IoU only for
// same-label candidates (~1/90 of elements touch the L2-resident box array).
__global__ void __launch_bounds__(TPB)
k_nms(const float* __restrict__ boxes,
      const float* __restrict__ raws,
      const float* __restrict__ sminit,
      unsigned long long* best,
      unsigned* cnt, unsigned* gen,
      float* bcast,                 // per image: box4, label, idx
      float* out) {
  extern __shared__ float s_sc[];               // CHUNK floats (90 KB)
  __shared__ unsigned long long s_red[TPB / 32];

  int img = blockIdx.x / BPI;
  int blk = blockIdx.x % BPI;
  int tid = threadIdx.x;
  size_t base = (size_t)img * M_ + (size_t)blk * CHUNK;

  // fill LDS score image asynchronously (ASYNCcnt path)
  for (int i = tid * 4; i + 3 < CHUNK; i += TPB * 4)
    async_load_f4(sminit + base + i, &s_sc[i]);
  wait_async0();
  __syncthreads();

  float* bc = bcast + img * 8;
  float4* s4 = (float4*)s_sc;

  for (int it = 0; it < DETS_; ++it) {
    float bbx1 = 0, bby1 = 0, bbx2 = 0, bby2 = 0, barea = 0;
    int bLab = -1, bIdx = -1;
    if (it > 0) {
      bbx1 = bc[0]; bby1 = bc[1]; bbx2 = bc[2]; bby2 = bc[3];
      bLab = (int)bc[4]; bIdx = (int)bc[5];
      barea = (bbx2 - bbx1) * (bby2 - bby1);
    }

    unsigned long long lbest = 0ull;
    for (int g = tid; g < GRP; g += TPB) {
      float4 sv = s4[g];                       // ds_load_b128
      float s[4] = {sv.x, sv.y, sv.z, sv.w};
      int j0 = blk * CHUNK + g * 4;
      if (it > 0) {
        bool dirty = false;
        int r0 = j0 % 90;                      // label-1 of first element
#pragma unroll
        for (int k = 0; k < 4; ++k) {
          if (s[k] > NEGH) {
            int j = j0 + k;
            int rk = r0 + k; rk = (rk >= 90) ? rk - 90 : rk;
            if (j == bIdx) {
              s[k] = NEGV; dirty = true;
            } else if (1 + rk == bLab) {
              float4 b = *(const float4*)(boxes + ((size_t)img * M_ + j) * 4);
              float ltx = fmaxf(bbx1, b.x), lty = fmaxf(bby1, b.y);
              float rbx = fminf(bbx2, b.z), rby = fminf(bby2, b.w);
              float iw = fmaxf(rbx - ltx, 0.0f), ih = fmaxf(rby - lty, 0.0f);
              float inter = iw * ih;
              float a2 = (b.z - b.x) * (b.w - b.y);
              float iou = inter / (barea + a2 - inter);
              if (iou > NMS_T_) { s[k] = NEGV; dirty = true; }
            }
          }
        }
        if (dirty) {
          sv.x = s[0]; sv.y = s[1]; sv.z = s[2]; sv.w = s[3];
          s4[g] = sv;                          // ds_store_b128 (rare)
        }
      }
      // pack: high = score bits (0 for suppressed), low = ~index (tie -> lowest idx)
#pragma unroll
      for (int k = 0; k < 4; ++k) {
        unsigned key = (s[k] > 0.0f) ? __float_as_uint(s[k]) : 0u;
        unsigned long long pk = ((unsigned long long)key << 32) |
                                (unsigned long long)(0xFFFFFFFFu - (unsigned)(j0 + k));
        lbest = (pk > lbest) ? pk : lbest;
      }
    }

    // wave32 reduction, then block, then image-wide atomicMax
    for (int off = 16; off; off >>= 1) {
      unsigned long long o = __shfl_xor(lbest, off, 32);
      lbest = (o > lbest) ? o : lbest;
    }
    if ((tid & 31) == 0) s_red[tid >> 5] = lbest;
    __syncthreads();
    if (tid == 0) {
      unsigned long long bb = s_red[0];
      for (int w = 1; w < TPB / 32; ++w) bb = (s_red[w] > bb) ? s_red[w] : bb;
      atomicMax(&best[img], bb);
    }

    image_barrier(cnt, gen, img);

    if (blk == 0 && tid == 0) {
      unsigned long long p =
          __hip_atomic_load(&best[img], __ATOMIC_RELAXED, __HIP_MEMORY_SCOPE_AGENT);
      unsigned key = (unsigned)(p >> 32);
      int idx = (int)(0xFFFFFFFFu - (unsigned)(p & 0xFFFFFFFFull));
      int ok = (key != 0u);
      size_t gi = (size_t)img * M_ + idx;
      float4 b = *(const float4*)(boxes + gi * 4);
      float sc = raws[gi];
      int lab = 1 + idx % 90;
      int o = img * DETS_ + it;
      float* ob  = out;                        // boxes  [B*DETS*4]
      float* os  = out + B_ * DETS_ * 4;       // scores [B*DETS]
      float* ol  = os + B_ * DETS_;            // labels [B*DETS]
      float* ook = ol + B_ * DETS_;            // keep_ok[B*DETS]
      ob[o * 4 + 0] = ok ? b.x : 0.0f;
      ob[o * 4 + 1] = ok ? b.y : 0.0f;
      ob[o * 4 + 2] = ok ? b.z : 0.0f;
      ob[o * 4 + 3] = ok ? b.w : 0.0f;
      os[o]  = ok ? sc : 0.0f;
      ol[o]  = ok ? (float)lab : 0.0f;
      ook[o] = ok ? 1.0f : 0.0f;
      bc[0] = b.x; bc[1] = b.y; bc[2] = b.z; bc[3] = b.w;
      bc[4] = (float)lab; bc[5] = (float)idx;
      __hip_atomic_store(&best[img], 0ull, __ATOMIC_RELAXED, __HIP_MEMORY_SCOPE_AGENT);
      __threadfence();
    }

    image_barrier(cnt, gen, img);
  }
}

// ------------- init: reset persistent-kernel control state each launch -------------
__global__ void k_init(unsigned long long* best, unsigned* cnt, unsigned* gen) {
  int i = threadIdx.x;
  if (i < B_) { best[i] = 0ull; cnt[i] = 0u; gen[i] = 0u; }
}

// ------------- phase 1: softmax + decode + clip + threshold -------------
// one wave32 per proposal row; lane c handles classes c, c+32, c+64
__global__ void __launch_bounds__(256)
k_decode(const float* __restrict__ logits,   // [B*N, 91]
         const float* __restrict__ breg,     // [B*N, 364]
         const float* __restrict__ props,    // [B, N, 4]
         float* __restrict__ boxes,          // [B*M, 4]
         float* __restrict__ raws,           // [B*M]
         float* __restrict__ sminit) {       // [B*M] score or NEG
  int gtid = blockIdx.x * blockDim.x + threadIdx.x;
  int row  = gtid >> 5;
  int lane = gtid & 31;
  if (row >= B_ * N_) return;
  int img = row / N_;
  int n   = row - img * N_;

  const float* lg = logits + (size_t)row * C_;
  float mx = -3.0e38f;
  for (int c = lane; c < C_; c += 32) mx = fmaxf(mx, lg[c]);
  for (int off = 16; off; off >>= 1) mx = fmaxf(mx, __shfl_xor(mx, off, 32));
  float sum = 0.0f;
  for (int c = lane; c < C_; c += 32) sum += expf(lg[c] - mx);
  for (int off = 16; off; off >>= 1) sum += __shfl_xor(sum, off, 32);
  float inv = 1.0f / sum;

  const float* p = props + ((size_t)img * N_ + n) * 4;
  float x1 = p[0], y1 = p[1], x2 = p[2], y2 = p[3];
  float pw = x2 - x1, ph = y2 - y1;
  float pcx = x1 + 0.5f * pw, pcy = y1 + 0.5f * ph;

  const float* rr = breg + (size_t)row * (C_ * 4);
  for (int c = lane; c < C_; c += 32) {
    if (c == 0) continue;                       // drop background
    float sc = expf(lg[c] - mx) * inv;
    float dx = rr[c * 4 + 0] / 10.0f;
    float dy = rr[c * 4 + 1] / 10.0f;
    float dw = fminf(rr[c * 4 + 2] / 5.0f, BBOX_CLIP);
    float dh = fminf(rr[c * 4 + 3] / 5.0f, BBOX_CLIP);
    float gcx = dx * pw + pcx, gcy = dy * ph + pcy;
    float gw = expf(dw) * pw,  gh = expf(dh) * ph;
    float bx1 = fminf(fmaxf(gcx - 0.5f * gw, 0.0f), WIMG);
    float by1 = fminf(fmaxf(gcy - 0.5f * gh, 0.0f), WIMG);
    float bx2 = fminf(fmaxf(gcx + 0.5f * gw, 0.0f), WIMG);
    float by2 = fminf(fmaxf(gcy + 0.5f * gh, 0.0f), WIMG);
    int j = n * 90 + (c - 1);
    size_t gi = (size_t)img * M_ + j;
    float4 bb; bb.x = bx1; bb.y = by1; bb.z = bx2; bb.w = by2;
    ((float4*)boxes)[gi] = bb;
    raws[gi] = sc;
    bool valid = (sc > SCORE_T_) && ((bx2 - bx1) >= MIN_SZ_) && ((by2 - by1) >= MIN_SZ_);
    sminit[gi] = valid ? sc : NEGV;
  }
}

extern "C" void kernel_launch(void* const* d_in, const int* in_sizes, int n_in,
                              void* d_out, int out_size, void* d_ws, size_t ws_size,
                              hipStream_t stream) {
  (void)in_sizes; (void)n_in; (void)out_size; (void)ws_size;
  const float* logits = (const float*)d_in[0];
  const float* breg   = (const float*)d_in[1];
  const float* props  = (const float*)d_in[2];
  float* out = (float*)d_out;

  // workspace layout (~34.6 MB)
  float* boxes  = (float*)d_ws;                                // B*M*4
  float* raws   = boxes + (size_t)B_ * M_ * 4;                 // B*M
  float* sminit = raws + (size_t)B_ * M_;                      // B*M
  unsigned long long* best = (unsigned long long*)(sminit + (size_t)B_ * M_); // [B]
  unsigned* cnt  = (unsigned*)(best + B_);                     // [B]
  unsigned* gen  = cnt + B_;                                   // [B]
  float* bcast   = (float*)(gen + B_);                         // [B*8]

  k_init<<<1, 64, 0, stream>>>(best, cnt, gen);
  k_decode<<<(B_ * N_ * 32) / 256, 256, 0, stream>>>(logits, breg, props,
                                                     boxes, raws, sminit);
  k_nms<<<B_ * BPI, TPB, CHUNK * sizeof(float), stream>>>(boxes, raws, sminit,
                                                          best, cnt, gen, bcast, out);
}